// GCN_67654324846730
// MI455X (gfx1250) — compile-verified
//
#include <hip/hip_runtime.h>
#include <hip/hip_bf16.h>
#include <math.h>

typedef __attribute__((ext_vector_type(2))) float v2f;
typedef __attribute__((ext_vector_type(4))) float v4f;
typedef __attribute__((ext_vector_type(8))) float v8f;

// ---------------------------------------------------------------------------
// float atomic add -> global_atomic_add_f32 (L2 atomic units)
// ---------------------------------------------------------------------------
__device__ __forceinline__ void atomAddF(float* p, float v) {
    unsafeAtomicAdd(p, v);
}

// ---------------------------------------------------------------------------
// pad x [N,3] -> x4 [N,4] (k=3 column zero) so layer-1 K is a multiple of 4
// ---------------------------------------------------------------------------
__global__ void gcn_pad_x(const float* __restrict__ x, float* __restrict__ x4, int nN) {
    int n = blockIdx.x * blockDim.x + threadIdx.x;
    if (n < nN) {
        v4f v;
        v.x = x[(size_t)n * 3 + 0];
        v.y = x[(size_t)n * 3 + 1];
        v.z = x[(size_t)n * 3 + 2];
        v.w = 0.0f;
        *(v4f*)(x4 + (size_t)n * 4) = v;
    }
}

// pad W4 [16,2] -> [16,16] zero-extended so the B-fragment loads are unguarded
__global__ void gcn_pad_w4(const float* __restrict__ W4, float* __restrict__ W4p) {
    int i = threadIdx.x;            // 256 threads
    int row = i >> 4, col = i & 15;
    W4p[i] = (col < 2) ? W4[row * 2 + col] : 0.0f;
}

// ---------------------------------------------------------------------------
// degree / dinv
// ---------------------------------------------------------------------------
__global__ void gcn_deg_init(float* __restrict__ deg, int nN) {
    int i = blockIdx.x * blockDim.x + threadIdx.x;
    if (i < nN) deg[i] = 1.0f;   // self-loop
}

__global__ void gcn_deg_count(const int* __restrict__ dst, float* __restrict__ deg, int nE) {
    int e = blockIdx.x * blockDim.x + threadIdx.x;
    if (e < nE) atomAddF(&deg[dst[e]], 1.0f);
}

__global__ void gcn_deg_rsqrt(float* __restrict__ deg, int nN) {
    int i = blockIdx.x * blockDim.x + threadIdx.x;
    if (i < nN) deg[i] = rsqrtf(deg[i]);   // deg >= 1 always (self-loops)
}

// ---------------------------------------------------------------------------
// GEMM: h = hin @ W via V_WMMA_F32_16X16X4_F32 (fp32, chained K=4 steps).
// Epilogue also writes agg = h * dinv^2 (self-loop message), so the scatter
// kernel needs no separate agg init.
//
// FIN % 4 == 0 (inputs pre-padded). WSTRIDE = row stride of W (W4 padded to 16).
// A-frag (16x4 f32): lane L holds row L%16, k = 2*(L/16)+j  -> one v2f load.
// B-frag (4x16 f32): lane L holds col L%16, k = 2*(L/16)+j.
// C/D (16x16 f32):   lane L, vgpr v -> row v + 8*(L/16), col L%16.
// Fast path (full 16-row tile): completely branch/guard-free.
// ---------------------------------------------------------------------------
template<int FIN, int FOUT, int WSTRIDE>
__global__ void gcn_gemm_selfloop(const float* __restrict__ hin,
                                  const float* __restrict__ W,
                                  const float* __restrict__ dinv,
                                  float* __restrict__ h,
                                  float* __restrict__ agg,
                                  int nN) {
    const int lane = threadIdx.x & 31;
    const int wave = threadIdx.x >> 5;
    const int tile = blockIdx.x * (blockDim.x >> 5) + wave;   // 16-row tile
    const int row0 = tile * 16;
    if (row0 >= nN) return;                                   // wave-uniform exit

    const int half = lane >> 4;   // 0 or 1
    const int l16  = lane & 15;

    constexpr int KSTEPS = FIN / 4;
    constexpr int NTILES = (FOUT + 15) / 16;
    const bool full = (row0 + 16 <= nN);

    // ---- A fragments: hoisted, reused across all column tiles ----
    v2f afrag[KSTEPS];
    if (full) {
        const float* ap = hin + (size_t)(row0 + l16) * FIN + half * 2;
        #pragma unroll
        for (int ks = 0; ks < KSTEPS; ++ks)
            afrag[ks] = *(const v2f*)(ap + ks * 4);
    } else {
        const int arow = row0 + l16;
        #pragma unroll
        for (int ks = 0; ks < KSTEPS; ++ks) {
            const int kb = ks * 4 + half * 2;
            v2f a;
            a.x = (arow < nN) ? hin[(size_t)arow * FIN + kb + 0] : 0.0f;
            a.y = (arow < nN) ? hin[(size_t)arow * FIN + kb + 1] : 0.0f;
            afrag[ks] = a;
        }
    }

    #pragma unroll
    for (int nt = 0; nt < NTILES; ++nt) {
        const int col0 = nt * 16;
        v8f acc = {};
        #pragma unroll
        for (int ks = 0; ks < KSTEPS; ++ks) {
            const int kb = ks * 4 + half * 2;
            v2f b;
            b.x = W[(size_t)(kb + 0) * WSTRIDE + col0 + l16];
            b.y = W[(size_t)(kb + 1) * WSTRIDE + col0 + l16];
            acc = __builtin_amdgcn_wmma_f32_16x16x4_f32(
                      false, afrag[ks], false, b, (short)0, acc, false, false);
        }
        if (full) {
            if (FOUT >= 16 || l16 < FOUT) {
                #pragma unroll
                for (int v = 0; v < 8; ++v) {
                    const int r = row0 + v + 8 * half;
                    const float di = dinv[r];
                    const size_t o = (size_t)r * FOUT + col0 + l16;
                    h[o]   = acc[v];
                    agg[o] = acc[v] * di * di;   // self-loop message
                }
            }
        } else {
            #pragma unroll
            for (int v = 0; v < 8; ++v) {
                const int r = row0 + v + 8 * half;
                const int c = col0 + l16;
                if (r < nN && c < FOUT) {
                    const float di = dinv[r];
                    const size_t o = (size_t)r * FOUT + c;
                    h[o]   = acc[v];
                    agg[o] = acc[v] * di * di;
                }
            }
        }
    }
}

// ---------------------------------------------------------------------------
// Edge scatter: agg[dst] += h[src] * dinv[src]*dinv[dst]
// One thread per (edge, 4-feature chunk); b128 gather + scalar f32 atomics.
// Feature arrays (<=12.8MB) are L2-resident on the 192MB L2.
// ---------------------------------------------------------------------------
template<int F>
__global__ void gcn_scatter(const int* __restrict__ src,
                            const int* __restrict__ dst,
                            const float* __restrict__ dinv,
                            const float* __restrict__ h,
                            float* __restrict__ agg,
                            int nE) {
    constexpr int CH  = (F >= 4) ? 4 : F;   // features per thread
    constexpr int TPE = F / CH;             // threads per edge
    int tid = blockIdx.x * blockDim.x + threadIdx.x;
    int e = tid / TPE;
    int c = tid - e * TPE;
    if (e >= nE) return;
    const int s = src[e];
    const int d = dst[e];
    const float norm = dinv[s] * dinv[d];
    const float* hp = h + (size_t)s * F + c * CH;
    float* ap = agg + (size_t)d * F + c * CH;
    if (CH == 4) {
        v4f m = *(const v4f*)hp;
        atomAddF(ap + 0, m.x * norm);
        atomAddF(ap + 1, m.y * norm);
        atomAddF(ap + 2, m.z * norm);
        atomAddF(ap + 3, m.w * norm);
    } else {
        #pragma unroll
        for (int i = 0; i < CH; ++i) atomAddF(ap + i, hp[i] * norm);
    }
}

// ---------------------------------------------------------------------------
// bias + ReLU (float4-vectorized):  out = relu(agg + b)
// ---------------------------------------------------------------------------
template<int F>
__global__ void gcn_bias_relu(const float* __restrict__ agg,
                              const float* __restrict__ b,
                              float* __restrict__ out,
                              int nN) {
    int idx = blockIdx.x * blockDim.x + threadIdx.x;   // per float4
    int base = idx * 4;
    if (base < nN * F) {
        int f = base & (F - 1);
        v4f v = *(const v4f*)(agg + base);
        v.x = fmaxf(v.x + b[f + 0], 0.0f);
        v.y = fmaxf(v.y + b[f + 1], 0.0f);
        v.z = fmaxf(v.z + b[f + 2], 0.0f);
        v.w = fmaxf(v.w + b[f + 3], 0.0f);
        *(v4f*)(out + base) = v;
    }
}

// ---------------------------------------------------------------------------
// pooling scratch zero
// ---------------------------------------------------------------------------
__global__ void gcn_pool_zero(float* __restrict__ pool, int n) {
    int i = blockIdx.x * blockDim.x + threadIdx.x;
    if (i < n) pool[i] = 0.0f;
}

// ---------------------------------------------------------------------------
// final layer bias (no relu) fused with mean-pool accumulation.
// LDS (ds_add_f32) reduction per block, then few global atomics per block.
// ---------------------------------------------------------------------------
__global__ void gcn_act4_pool(const float* __restrict__ agg,   // N x 2
                              const float* __restrict__ b4,    // 2
                              const int* __restrict__ batch,   // N
                              float* __restrict__ poolsum,     // G*2
                              float* __restrict__ poolcnt,     // G
                              int nN, int nG) {
    __shared__ float ssum[128];   // G<=64 graphs x 2
    __shared__ float scnt[64];
    if (threadIdx.x < 128) ssum[threadIdx.x] = 0.0f;
    if (threadIdx.x < 64)  scnt[threadIdx.x] = 0.0f;
    __syncthreads();
    int n = blockIdx.x * blockDim.x + threadIdx.x;
    if (n < nN) {
        float v0 = agg[(size_t)n * 2 + 0] + b4[0];
        float v1 = agg[(size_t)n * 2 + 1] + b4[1];
        int g = batch[n];
        atomicAdd(&ssum[g * 2 + 0], v0);   // ds_add_f32
        atomicAdd(&ssum[g * 2 + 1], v1);
        atomicAdd(&scnt[g], 1.0f);
    }
    __syncthreads();
    if (threadIdx.x < 128 && (int)(threadIdx.x >> 1) < nG) {
        float v = ssum[threadIdx.x];
        if (v != 0.0f) atomAddF(&poolsum[threadIdx.x], v);
    }
    if (threadIdx.x < 64 && (int)threadIdx.x < nG) {
        float v = scnt[threadIdx.x];
        if (v != 0.0f) atomAddF(&poolcnt[threadIdx.x], v);
    }
}

// ---------------------------------------------------------------------------
// log_softmax over 2 classes per graph
// ---------------------------------------------------------------------------
__global__ void gcn_logsoftmax(const float* __restrict__ poolsum,
                               const float* __restrict__ poolcnt,
                               float* __restrict__ out, int nG) {
    int g = blockIdx.x * blockDim.x + threadIdx.x;
    if (g < nG) {
        float c = fmaxf(poolcnt[g], 1.0f);
        float a = poolsum[g * 2 + 0] / c;
        float b = poolsum[g * 2 + 1] / c;
        float m = fmaxf(a, b);
        float lse = m + logf(expf(a - m) + expf(b - m));
        out[g * 2 + 0] = a - lse;
        out[g * 2 + 1] = b - lse;
    }
}

// ---------------------------------------------------------------------------
// host launcher
// ---------------------------------------------------------------------------
extern "C" void kernel_launch(void* const* d_in, const int* in_sizes, int n_in,
                              void* d_out, int out_size, void* d_ws, size_t ws_size,
                              hipStream_t stream) {
    const float* x     = (const float*)d_in[0];
    const int*   eidx  = (const int*)d_in[1];
    const int*   batch = (const int*)d_in[2];
    const float* W1 = (const float*)d_in[3];
    const float* b1 = (const float*)d_in[4];
    const float* W2 = (const float*)d_in[5];
    const float* b2 = (const float*)d_in[6];
    const float* W3 = (const float*)d_in[7];
    const float* b3 = (const float*)d_in[8];
    const float* W4 = (const float*)d_in[9];
    const float* b4 = (const float*)d_in[10];

    const int N = in_sizes[0] / 3;
    const int E = in_sizes[1] / 2;
    const int G = out_size / 2;

    const int* src = eidx;       // edge_index[0]
    const int* dst = eidx + E;   // edge_index[1]

    // workspace layout (floats), 1KB-aligned segments
    const size_t Npad = ((size_t)N + 255) & ~(size_t)255;
    float* ws    = (float*)d_ws;
    float* dinv  = ws;                    // Npad
    float* h     = dinv + Npad;           // Npad*32
    float* agg   = h    + Npad * 32;      // Npad*32
    float* hin   = agg  + Npad * 32;      // Npad*32 (also holds padded x4)
    float* pools = hin  + Npad * 32;      // G*2
    float* poolc = pools + 2 * G;         // G
    float* W4pad = poolc + ((G + 255) & ~255); // 256 floats
    (void)ws_size; (void)n_in;

    const int T = 256;
    const int nTiles  = (N + 15) / 16;
    const int gemmBlk = (nTiles + 7) / 8;   // 8 waves / block

    // input padding + weight padding (no cross-deps, run up front)
    gcn_pad_x <<<(N + T - 1) / T, T, 0, stream>>>(x, hin, N);
    gcn_pad_w4<<<1, 256, 0, stream>>>(W4, W4pad);

    // degrees -> dinv
    gcn_deg_init <<<(N + T - 1) / T, T, 0, stream>>>(dinv, N);
    gcn_deg_count<<<(E + T - 1) / T, T, 0, stream>>>(dst, dinv, E);
    gcn_deg_rsqrt<<<(N + T - 1) / T, T, 0, stream>>>(dinv, N);

    // ---- layer 1: (padded) 4 -> 16, ReLU
    gcn_gemm_selfloop<4, 16, 16><<<gemmBlk, T, 0, stream>>>(hin, W1, dinv, h, agg, N);
    gcn_scatter<16><<<(E * 4 + T - 1) / T, T, 0, stream>>>(src, dst, dinv, h, agg, E);
    gcn_bias_relu<16><<<((N * 16) / 4 + T - 1) / T, T, 0, stream>>>(agg, b1, hin, N);

    // ---- layer 2: 16 -> 32, ReLU
    gcn_gemm_selfloop<16, 32, 32><<<gemmBlk, T, 0, stream>>>(hin, W2, dinv, h, agg, N);
    gcn_scatter<32><<<(E * 8 + T - 1) / T, T, 0, stream>>>(src, dst, dinv, h, agg, E);
    gcn_bias_relu<32><<<((N * 32) / 4 + T - 1) / T, T, 0, stream>>>(agg, b2, hin, N);

    // ---- layer 3: 32 -> 16, ReLU
    gcn_gemm_selfloop<32, 16, 16><<<gemmBlk, T, 0, stream>>>(hin, W3, dinv, h, agg, N);
    gcn_scatter<16><<<(E * 4 + T - 1) / T, T, 0, stream>>>(src, dst, dinv, h, agg, E);
    gcn_bias_relu<16><<<((N * 16) / 4 + T - 1) / T, T, 0, stream>>>(agg, b3, hin, N);

    // ---- layer 4: 16 -> 2 (padded weights, no ReLU), then pool + log_softmax
    gcn_gemm_selfloop<16, 2, 16><<<gemmBlk, T, 0, stream>>>(hin, W4pad, dinv, h, agg, N);
    gcn_scatter<2><<<(E + T - 1) / T, T, 0, stream>>>(src, dst, dinv, h, agg, E);

    gcn_pool_zero<<<1, T, 0, stream>>>(pools, 3 * G);   // pools + poolc contiguous
    gcn_act4_pool<<<(N + T - 1) / T, T, 0, stream>>>(agg, b4, batch, pools, poolc, N, G);
    gcn_logsoftmax<<<1, T, 0, stream>>>(pools, poolc, (float*)d_out, G);
}